// GAT_10368051053146
// MI455X (gfx1250) — compile-verified
//
#include <hip/hip_runtime.h>
#include <math.h>

typedef float v2f __attribute__((ext_vector_type(2)));
typedef float v8f __attribute__((ext_vector_type(8)));

#define N_NODES 50000
#define HIDDEN  128
#define E_RAW   400000
#define E_TOT   (E_RAW + N_NODES)   // self loops appended
#define NEG_SLOPE 0.2f

// ---------------------------------------------------------------------------
// Dense projection H = X @ W using V_WMMA_F32_16X16X4_F32 (fp32, no downcast).
// Block = 256 threads = 8 waves. Block owns a 16-row tile of X (staged in LDS,
// 8 KB); wave w owns 16-col tile w of W. K-loop: 32 WMMAs of 16x16x4.
// Fragment layouts per ISA 7.12.2 (32-bit A 16x4 / B 4x16 / C 16x16):
//   A: a[v] = X[rowb + (lane&15)][k0 + 2*(lane>>4) + v]
//   B: b[v] = W[k0 + 2*(lane>>4) + v][colb + (lane&15)]
//   C: c[v] -> H[rowb + v + 8*(lane>>4)][colb + (lane&15)]
// ---------------------------------------------------------------------------
__global__ void __launch_bounds__(256)
gemm_wmma(const float* __restrict__ X, const float* __restrict__ W,
          float* __restrict__ H) {
    __shared__ float ldsA[16 * HIDDEN];
    const int rowb = blockIdx.x * 16;

    for (int i = threadIdx.x; i < 16 * HIDDEN; i += blockDim.x)
        ldsA[i] = X[rowb * HIDDEN + i];
    __syncthreads();

    const int wave = threadIdx.x >> 5;
    const int lane = threadIdx.x & 31;
    const int m    = lane & 15;
    const int khi  = lane >> 4;      // 0 or 1
    const int colb = wave * 16;

    v8f c = {};
#pragma unroll
    for (int k0 = 0; k0 < HIDDEN; k0 += 4) {
        const int k = k0 + 2 * khi;
        v2f a;
        a.x = ldsA[m * HIDDEN + k];
        a.y = ldsA[m * HIDDEN + k + 1];
        v2f b;
        b.x = W[k * HIDDEN + colb + m];
        b.y = W[(k + 1) * HIDDEN + colb + m];
        c = __builtin_amdgcn_wmma_f32_16x16x4_f32(
                /*neg_a=*/false, a, /*neg_b=*/false, b,
                /*c_mod=*/(short)0, c, /*reuse_a=*/false, /*reuse_b=*/false);
    }

#pragma unroll
    for (int v = 0; v < 8; ++v)
        H[(rowb + v + 8 * khi) * HIDDEN + colb + m] = c[v];
}

// ---------------------------------------------------------------------------
// Per-node attention coefficients: as[n][h] = sum_c H[n][h*ch+c]*a_src[h][c]
// ---------------------------------------------------------------------------
__global__ void alpha_kernel(const float* __restrict__ H,
                             const float* __restrict__ a_src,
                             const float* __restrict__ a_dst,
                             float* __restrict__ as_, float* __restrict__ ad_,
                             int heads) {
    const int n = blockIdx.x * blockDim.x + threadIdx.x;
    if (n >= N_NODES) return;
    const int ch = HIDDEN / heads;
    for (int h = 0; h < heads; ++h) {
        float s = 0.f, d = 0.f;
        for (int c = 0; c < ch; ++c) {
            const float hv = H[n * HIDDEN + h * ch + c];
            s += hv * a_src[h * ch + c];
            d += hv * a_dst[h * ch + c];
        }
        as_[n * heads + h] = s;
        ad_[n * heads + h] = d;
    }
}

__global__ void init_kernel(float* __restrict__ accum,
                            float* __restrict__ emax,
                            float* __restrict__ denom, int heads) {
    const int i = blockIdx.x * blockDim.x + threadIdx.x;
    if (i < N_NODES * HIDDEN) accum[i] = 0.f;
    if (i < N_NODES * heads) { emax[i] = -3.0e38f; denom[i] = 0.f; }
}

__device__ __forceinline__ int edge_src(const int* __restrict__ ei, int e) {
    return (e < E_RAW) ? ei[e] : (e - E_RAW);
}
__device__ __forceinline__ int edge_dst(const int* __restrict__ ei, int e) {
    return (e < E_RAW) ? ei[E_RAW + e] : (e - E_RAW);
}
__device__ __forceinline__ float lrelu(float v) {
    return v > 0.f ? v : NEG_SLOPE * v;
}
// fp32 atomic max via IEEE int-ordering (maps onto native integer atomics).
__device__ __forceinline__ void atomicMaxF(float* addr, float v) {
    if (v >= 0.f)
        atomicMax((int*)addr, __float_as_int(v));
    else
        atomicMin((unsigned int*)addr, (unsigned int)__float_as_int(v));
}

// Pass A: segment max over dst
__global__ void edge_max(const int* __restrict__ ei,
                         const float* __restrict__ as_,
                         const float* __restrict__ ad_,
                         float* __restrict__ emax, int heads) {
    const int e = blockIdx.x * blockDim.x + threadIdx.x;
    if (e >= E_TOT) return;
    const int s = edge_src(ei, e), d = edge_dst(ei, e);
    for (int h = 0; h < heads; ++h) {
        const float v = lrelu(as_[s * heads + h] + ad_[d * heads + h]);
        atomicMaxF(&emax[d * heads + h], v);
    }
}

// Pass B: segment sum of exp(e - max)
__global__ void edge_expsum(const int* __restrict__ ei,
                            const float* __restrict__ as_,
                            const float* __restrict__ ad_,
                            const float* __restrict__ emax,
                            float* __restrict__ denom, int heads) {
    const int e = blockIdx.x * blockDim.x + threadIdx.x;
    if (e >= E_TOT) return;
    const int s = edge_src(ei, e), d = edge_dst(ei, e);
    for (int h = 0; h < heads; ++h) {
        const float v = lrelu(as_[s * heads + h] + ad_[d * heads + h]);
        atomicAdd(&denom[d * heads + h], expf(v - emax[d * heads + h]));
    }
}

// Pass C: one block (128 threads = channels) per edge; coalesced h[src] row
// gather (stays in 192MB L2), fp32 atomic scatter to accum[dst].
__global__ void __launch_bounds__(HIDDEN)
edge_scatter(const int* __restrict__ ei, const float* __restrict__ H,
             const float* __restrict__ as_, const float* __restrict__ ad_,
             const float* __restrict__ emax, const float* __restrict__ denom,
             float* __restrict__ accum, int heads) {
    const int e = blockIdx.x;
    const int c = threadIdx.x;
    const int s = edge_src(ei, e), d = edge_dst(ei, e);
    const int ch = HIDDEN / heads;
    const int h  = c / ch;
    const float v     = lrelu(as_[s * heads + h] + ad_[d * heads + h]);
    const float alpha = expf(v - emax[d * heads + h]) / denom[d * heads + h];
    atomicAdd(&accum[d * HIDDEN + c], H[s * HIDDEN + c] * alpha);
}

__global__ void finalize_kernel(const float* __restrict__ accum,
                                const float* __restrict__ b,
                                float* __restrict__ out, int do_relu) {
    const int i = blockIdx.x * blockDim.x + threadIdx.x;
    if (i >= N_NODES * HIDDEN) return;
    float v = accum[i] + b[i & (HIDDEN - 1)];
    if (do_relu) v = v > 0.f ? v : 0.f;
    out[i] = v;
}

// ---------------------------------------------------------------------------
static void run_layer(const float* x_in, const int* ei, const float* W,
                      const float* a_src, const float* a_dst, const float* b,
                      int heads, int do_relu, float* out,
                      float* H, float* accum, float* as_, float* ad_,
                      float* emax, float* denom, hipStream_t stream) {
    gemm_wmma<<<N_NODES / 16, 256, 0, stream>>>(x_in, W, H);
    alpha_kernel<<<(N_NODES + 255) / 256, 256, 0, stream>>>(H, a_src, a_dst,
                                                            as_, ad_, heads);
    init_kernel<<<(N_NODES * HIDDEN + 255) / 256, 256, 0, stream>>>(accum, emax,
                                                                    denom, heads);
    edge_max<<<(E_TOT + 255) / 256, 256, 0, stream>>>(ei, as_, ad_, emax, heads);
    edge_expsum<<<(E_TOT + 255) / 256, 256, 0, stream>>>(ei, as_, ad_, emax,
                                                         denom, heads);
    edge_scatter<<<E_TOT, HIDDEN, 0, stream>>>(ei, H, as_, ad_, emax, denom,
                                               accum, heads);
    finalize_kernel<<<(N_NODES * HIDDEN + 255) / 256, 256, 0, stream>>>(
        accum, b, out, do_relu);
}

extern "C" void kernel_launch(void* const* d_in, const int* in_sizes, int n_in,
                              void* d_out, int out_size, void* d_ws,
                              size_t ws_size, hipStream_t stream) {
    const float* x      = (const float*)d_in[0];
    const int*   ei     = (const int*)d_in[1];
    const float* W1     = (const float*)d_in[2];
    const float* a_src1 = (const float*)d_in[3];
    const float* a_dst1 = (const float*)d_in[4];
    const float* b1     = (const float*)d_in[5];
    const float* W2     = (const float*)d_in[6];
    const float* a_src2 = (const float*)d_in[7];
    const float* a_dst2 = (const float*)d_in[8];
    const float* b2     = (const float*)d_in[9];
    const float* W3     = (const float*)d_in[10];
    const float* a_src3 = (const float*)d_in[11];
    const float* a_dst3 = (const float*)d_in[12];
    const float* b3     = (const float*)d_in[13];
    float* out = (float*)d_out;

    float* ws    = (float*)d_ws;
    float* H     = ws;                          // 50000*128
    float* accum = H     + (size_t)N_NODES * HIDDEN;
    float* act   = accum + (size_t)N_NODES * HIDDEN;
    float* as_   = act   + (size_t)N_NODES * HIDDEN;   // 50000*4
    float* ad_   = as_   + (size_t)N_NODES * 4;
    float* emax  = ad_   + (size_t)N_NODES * 4;
    float* denom = emax  + (size_t)N_NODES * 4;

    run_layer(x,   ei, W1, a_src1, a_dst1, b1, 4, 1, act, H, accum, as_, ad_,
              emax, denom, stream);
    run_layer(act, ei, W2, a_src2, a_dst2, b2, 4, 1, act, H, accum, as_, ad_,
              emax, denom, stream);
    run_layer(act, ei, W3, a_src3, a_dst3, b3, 1, 0, out, H, accum, as_, ad_,
              emax, denom, stream);
}